// GlobalItemConv_36077725286627
// MI455X (gfx1250) — compile-verified
//
#include <hip/hip_runtime.h>
#include <hip/hip_bf16.h>

#define EMB 100
#define EPS 1e-12f

typedef float v2f __attribute__((ext_vector_type(2)));
typedef float v8f __attribute__((ext_vector_type(8)));

// -------------------------------------------------------------------------
// Kernel 1: h = x @ W^T  via V_WMMA_F32_16X16X4_F32, one wave per 16x16 tile.
// A-tile (16x4 of x):  lane l -> row m0+(l&15), K = k + 2*(l>>4) + {0,1}
// B-tile (4x16 of W^T): vgpr v, lane l -> K = k + 2*(l>>4)+v, N = o0+(l&15)
//   B[K][N] = W^T[K][N] = W[N][K], so lane l loads W[(o0+(l&15))*100 + k+2h +{0,1}]
// C/D: vgpr v, lane l -> row m0 + v + 8*(l>>4), col o0 + (l&15)
//
// Out-of-range columns (o0+lm >= 100, only in the last N-tile): a B lane only
// influences its own D column, and each lane stores only its own column, so we
// clamp the W row pointer to a valid row and simply skip the store. No branches
// or zero-fills in the inner loop.
// -------------------------------------------------------------------------
__global__ __launch_bounds__(256)
void gic_gemm_wmma(const float* __restrict__ x, const float* __restrict__ W,
                   float* __restrict__ h, int nTilesN, int nWaves) {
    const int waveId = (blockIdx.x * blockDim.x + threadIdx.x) >> 5;
    if (waveId >= nWaves) return;
    const int lane = threadIdx.x & 31;
    const int mTile = waveId / nTilesN;
    const int nTile = waveId % nTilesN;

    const int m0 = mTile * 16;
    const int o0 = nTile * 16;
    const int lm = lane & 15;
    const int half = lane >> 4;

    const int aRow = m0 + lm;                  // row of x this lane feeds
    const int bColRaw = o0 + lm;               // output column (row of W)
    const bool bValid = (bColRaw < EMB);
    const int bCol = bValid ? bColRaw : (EMB - 1);   // clamp: garbage lands in
                                                     // unstored D columns only

    const float* __restrict__ aPtr = x + (long)aRow * EMB + 2 * half;
    const float* __restrict__ bPtr = W + (long)bCol * EMB + 2 * half;

    v8f c = {0.f, 0.f, 0.f, 0.f, 0.f, 0.f, 0.f, 0.f};

    // K = 100 -> 25 steps of K=4; branch-free inner loop
    #pragma unroll
    for (int k = 0; k < EMB; k += 4) {
        v2f a = *(const v2f*)(aPtr + k);       // 8B aligned
        v2f b = *(const v2f*)(bPtr + k);       // 8B aligned
        c = __builtin_amdgcn_wmma_f32_16x16x4_f32(
                /*neg_a=*/false, a, /*neg_b=*/false, b,
                /*c_mod=*/(short)0, c, /*reuse_a=*/false, /*reuse_b=*/false);
    }

    if (bValid) {
        #pragma unroll
        for (int v = 0; v < 8; ++v) {
            int row = m0 + v + 8 * half;
            h[(long)row * EMB + bColRaw] = c[v];
        }
    }
}

// -------------------------------------------------------------------------
// Kernel 2: scatter-add. One wave per edge; lanes stride the 100 features.
// out[dst[e], j] += ew[e] * h[src[e], j]
// -------------------------------------------------------------------------
__global__ __launch_bounds__(256)
void gic_scatter(const float* __restrict__ h, const float* __restrict__ ew,
                 const int* __restrict__ dst, const int* __restrict__ src,
                 float* __restrict__ out, int nEdges) {
    const int e = (blockIdx.x * blockDim.x + threadIdx.x) >> 5;
    if (e >= nEdges) return;
    const int lane = threadIdx.x & 31;

    const int s = src[e];
    const int d = dst[e];
    const float w = ew[e];

    const float* __restrict__ hrow = h + (long)s * EMB;
    float* __restrict__ orow = out + (long)d * EMB;

    #pragma unroll
    for (int j = lane; j < EMB; j += 32) {
        atomicAdd(&orow[j], w * hrow[j]);
    }
}

// -------------------------------------------------------------------------
// Kernel 3: row-wise L2 normalize in place. One wave per row.
// -------------------------------------------------------------------------
__global__ __launch_bounds__(256)
void gic_normalize(float* __restrict__ out, int nRows) {
    const int row = (blockIdx.x * blockDim.x + threadIdx.x) >> 5;
    if (row >= nRows) return;
    const int lane = threadIdx.x & 31;

    float* __restrict__ r = out + (long)row * EMB;

    float v0 = r[lane];
    float v1 = r[lane + 32];
    float v2 = r[lane + 64];
    float v3 = (lane < EMB - 96) ? r[lane + 96] : 0.f;

    float ss = v0 * v0 + v1 * v1 + v2 * v2 + v3 * v3;
    // wave32 butterfly reduction
    #pragma unroll
    for (int m = 16; m >= 1; m >>= 1)
        ss += __shfl_xor(ss, m, 32);

    const float norm = sqrtf(ss);
    const float scale = 1.0f / fmaxf(norm, EPS);

    r[lane]      = v0 * scale;
    r[lane + 32] = v1 * scale;
    r[lane + 64] = v2 * scale;
    if (lane < EMB - 96) r[lane + 96] = v3 * scale;
}

extern "C" void kernel_launch(void* const* d_in, const int* in_sizes, int n_in,
                              void* d_out, int out_size, void* d_ws, size_t ws_size,
                              hipStream_t stream) {
    const float* x  = (const float*)d_in[0];   // [N, 100]
    const float* W  = (const float*)d_in[1];   // [100, 100] (torch [out,in])
    const float* ew = (const float*)d_in[2];   // [E]
    const int*   ei = (const int*)d_in[3];     // [2, E]; row0 = dst, row1 = src

    const int N = in_sizes[0] / EMB;           // 50000
    const int E = in_sizes[2];                 // 800000
    const int* dst = ei;
    const int* src = ei + E;

    float* h   = (float*)d_ws;                 // [N, 100] intermediate (20 MB)
    float* out = (float*)d_out;                // [N, 100]

    // segment_sum accumulates from zero; harness does not re-zero between replays
    hipMemsetAsync(out, 0, (size_t)N * EMB * sizeof(float), stream);

    // GEMM: one wave per 16x16 tile
    const int nTilesM = (N + 15) / 16;         // 3125
    const int nTilesN = (EMB + 15) / 16;       // 7
    const int nWaves  = nTilesM * nTilesN;     // 21875
    {
        const int threads = 256;               // 8 waves / block
        const int blocks  = (nWaves * 32 + threads - 1) / threads;
        gic_gemm_wmma<<<blocks, threads, 0, stream>>>(x, W, h, nTilesN, nWaves);
    }

    // Scatter-add: one wave per edge
    {
        const int threads = 256;
        const long totalThreads = (long)E * 32;
        const int blocks = (int)((totalThreads + threads - 1) / threads);
        gic_scatter<<<blocks, threads, 0, stream>>>(h, ew, dst, src, out, E);
    }

    // Normalize: one wave per row
    {
        const int threads = 256;
        const long totalThreads = (long)N * 32;
        const int blocks = (int)((totalThreads + threads - 1) / threads);
        gic_normalize<<<blocks, threads, 0, stream>>>(out, N);
    }
}